// Sparsemax_29858612642052
// MI455X (gfx1250) — compile-verified
//
#include <hip/hip_runtime.h>
#include <stdint.h>

// ---------------------------------------------------------------------------
// Sparsemax over rows of 64 fp32 values (N = 1048576 rows).
// Memory-bound (512 MB traffic -> ~22us @ 23.3 TB/s). Sort-free Michelot
// fixed-point for tau; gfx1250 async global->LDS staging + ds_swizzle quad
// reductions; NT coalesced stores.
// ---------------------------------------------------------------------------

typedef float v4f __attribute__((ext_vector_type(4)));
typedef int   v4i __attribute__((ext_vector_type(4)));

#define LDSF 68   // floats per padded LDS row (272 B) -> conflict-free b128

// ds_swizzle group-of-32 XOR patterns: offset = (xor<<10) | and_mask(0x1f)
#define SWZ_X1  0x041F
#define SWZ_X2  0x081F
#define SWZ_X4  0x101F
#define SWZ_X8  0x201F
#define SWZ_X16 0x401F

template <int IMM>
__device__ __forceinline__ float swz(float x) {
  return __int_as_float(__builtin_amdgcn_ds_swizzle(__float_as_int(x), IMM));
}

__device__ __forceinline__ unsigned ballot32(bool p) {
#if __has_builtin(__builtin_amdgcn_ballot_w32)
  return __builtin_amdgcn_ballot_w32(p);
#else
  return (unsigned)__ballot(p);
#endif
}

// Async copy 16 bytes global -> LDS (gfx1250 GLOBAL_LOAD_ASYNC_TO_LDS_B128).
__device__ __forceinline__ void async_g2l_b128(float* lptr, const float* gptr) {
#if __has_builtin(__builtin_amdgcn_global_load_async_to_lds_b128)
  __builtin_amdgcn_global_load_async_to_lds_b128(
      (__attribute__((address_space(1))) v4i*)gptr,
      (__attribute__((address_space(3))) v4i*)lptr,
      0 /*imm offset*/, 0 /*cpol*/);
#else
  uint32_t l = (uint32_t)(uintptr_t)lptr;          // low 32 bits == LDS offset
  uint64_t g = (uint64_t)(uintptr_t)gptr;
  asm volatile("global_load_async_to_lds_b128 %0, %1, off"
               :: "v"(l), "v"(g) : "memory");
#endif
}

__device__ __forceinline__ void wait_async0() {
#if __has_builtin(__builtin_amdgcn_s_wait_asynccnt)
  __builtin_amdgcn_s_wait_asynccnt(0);
#else
  asm volatile("s_wait_asynccnt 0" ::: "memory");
#endif
  asm volatile("" ::: "memory");  // keep LDS reads below the wait
}

// ---------------------------------------------------------------------------
// Fast path: 256 threads = 8 waves; wave handles 8 rows; lane owns 16 floats.
// Requires row count to be a multiple of 64 per grid sizing (guaranteed by
// the launcher; tail rows go to the generic kernel below).
// ---------------------------------------------------------------------------
__global__ __launch_bounds__(256)
void sparsemax_tile_kernel(const float* __restrict__ x, float* __restrict__ out) {
  __shared__ float tile[64 * LDSF];               // 17408 B

  const int tid  = threadIdx.x;
  const int wave = tid >> 5;
  const int lane = tid & 31;

  const size_t waveRow = (size_t)blockIdx.x * 64u + (size_t)wave * 8u;
  const float* gsrc = x   + waveRow * 64u;
  float*       gdst = out + waveRow * 64u;
  float* lds = &tile[(wave * 8) * LDSF];

  // ---- stage 8 rows (2 KiB) global -> padded LDS via async b128 copies ----
  // global side: chunk g = j*32+lane  -> gsrc + g*16B (fully coalesced)
  // LDS side:    row = g>>4, c = g&15 -> row*272 + c*16 (conflict-free)
#pragma unroll
  for (int j = 0; j < 4; ++j) {
    const int g   = j * 32 + lane;
    const int row = g >> 4;
    const int c   = g & 15;
    async_g2l_b128(&lds[row * LDSF + c * 4], gsrc + (size_t)g * 4u);
  }
  wait_async0();   // same wave produced & consumes -> no block barrier needed

  // ---- lane's 16 contiguous elements of its row ----
  const int rl = lane >> 2;                       // row in wave: 0..7
  const int q  = lane & 3;                        // quarter of row
  float* lrow = &lds[rl * LDSF + q * 16];

  float v[16];
#pragma unroll
  for (int k = 0; k < 4; ++k) {
    v4f t4 = *(const v4f*)(lrow + k * 4);
    v[4*k+0] = t4.x; v[4*k+1] = t4.y; v[4*k+2] = t4.z; v[4*k+3] = t4.w;
  }

  // ---- row max (seed: tau >= max-1 always) ----
  float m = v[0];
#pragma unroll
  for (int i = 1; i < 16; ++i) m = fmaxf(m, v[i]);
  m = fmaxf(m, swz<SWZ_X1>(m));
  m = fmaxf(m, swz<SWZ_X2>(m));

  // ---- Michelot fixed point: t -> (sum_{x>t} x - 1) / count_{x>t} ----
  float t = m - 1.0f;
  float s = 0.0f, c = 1.0f;
#pragma unroll 1
  for (int it = 0; it < 64; ++it) {
    s = 0.0f; c = 0.0f;
#pragma unroll
    for (int i = 0; i < 16; ++i) {
      const float ai = (v[i] > t) ? 1.0f : 0.0f;
      c += ai;
      s = __builtin_fmaf(ai, v[i], s);
    }
    s += swz<SWZ_X1>(s);  c += swz<SWZ_X1>(c);
    s += swz<SWZ_X2>(s);  c += swz<SWZ_X2>(c);
    const float tn = (s - 1.0f) / c;
    const unsigned nd = ballot32(tn != t);        // wave-uniform decision
    t = tn;
    if (nd == 0u) break;                          // all 8 rows converged
  }
  // Reference tau has a double "-1": tau = (S_rho - 2) / rho
  const float tau = (s - 2.0f) / c;

  // ---- results back through padded LDS, then coalesced NT b128 stores ----
#pragma unroll
  for (int k = 0; k < 4; ++k) {
    v4f r;
    r.x = fmaxf(0.0f, v[4*k+0] - tau);
    r.y = fmaxf(0.0f, v[4*k+1] - tau);
    r.z = fmaxf(0.0f, v[4*k+2] - tau);
    r.w = fmaxf(0.0f, v[4*k+3] - tau);
    *(v4f*)(lrow + k * 4) = r;                    // ds_store_b128
  }
#pragma unroll
  for (int j = 0; j < 4; ++j) {
    const int g   = j * 32 + lane;
    const int row = g >> 4;
    const int c2  = g & 15;
    v4f val = *(const v4f*)&lds[row * LDSF + c2 * 4];   // ds_load_b128
    __builtin_nontemporal_store(val, (v4f*)(gdst + (size_t)g * 4u));
  }
}

// ---------------------------------------------------------------------------
// Generic path (any row count): one wave per row, 2 elements per lane.
// ---------------------------------------------------------------------------
__global__ __launch_bounds__(256)
void sparsemax_row_kernel(const float* __restrict__ x, float* __restrict__ out,
                          long long firstRow, long long nRows) {
  const int lane = threadIdx.x & 31;
  const int wIdx = blockIdx.x * (blockDim.x >> 5) + (threadIdx.x >> 5);
  const int nW   = gridDim.x * (blockDim.x >> 5);

  for (long long r = firstRow + wIdx; r < firstRow + nRows; r += nW) {
    const float* px = x + r * 64;
    const float a = px[lane];
    const float b = px[lane + 32];

    float m = fmaxf(a, b);
    m = fmaxf(m, swz<SWZ_X1>(m));
    m = fmaxf(m, swz<SWZ_X2>(m));
    m = fmaxf(m, swz<SWZ_X4>(m));
    m = fmaxf(m, swz<SWZ_X8>(m));
    m = fmaxf(m, swz<SWZ_X16>(m));

    float t = m - 1.0f, s = 0.0f, c = 1.0f;
#pragma unroll 1
    for (int it = 0; it < 64; ++it) {
      const float aa = (a > t) ? 1.0f : 0.0f;
      const float bb = (b > t) ? 1.0f : 0.0f;
      c = aa + bb;
      s = __builtin_fmaf(aa, a, bb * b);
      s += swz<SWZ_X1>(s);  c += swz<SWZ_X1>(c);
      s += swz<SWZ_X2>(s);  c += swz<SWZ_X2>(c);
      s += swz<SWZ_X4>(s);  c += swz<SWZ_X4>(c);
      s += swz<SWZ_X8>(s);  c += swz<SWZ_X8>(c);
      s += swz<SWZ_X16>(s); c += swz<SWZ_X16>(c);
      const float tn = (s - 1.0f) / c;
      const unsigned nd = ballot32(tn != t);
      t = tn;
      if (nd == 0u) break;
    }
    const float tau = (s - 2.0f) / c;

    float* po = out + r * 64;
    po[lane]      = fmaxf(0.0f, a - tau);
    po[lane + 32] = fmaxf(0.0f, b - tau);
  }
}

// ---------------------------------------------------------------------------
extern "C" void kernel_launch(void* const* d_in, const int* in_sizes, int n_in,
                              void* d_out, int out_size, void* d_ws, size_t ws_size,
                              hipStream_t stream) {
  (void)n_in; (void)out_size; (void)d_ws; (void)ws_size;
  const float* x = (const float*)d_in[0];
  float* out = (float*)d_out;

  const long long elems = (long long)in_sizes[0];
  const long long rows  = elems / 64;                 // 64 experts per row
  const long long fullBlocks = rows / 64;             // 64 rows per block
  const long long tail       = rows - fullBlocks * 64;

  if (fullBlocks > 0) {
    sparsemax_tile_kernel<<<(unsigned)fullBlocks, 256, 0, stream>>>(x, out);
  }
  if (tail > 0) {
    const int blocks2 = (int)((tail + 7) / 8);        // 8 waves per block
    sparsemax_row_kernel<<<blocks2, 256, 0, stream>>>(x, out,
                                                      fullBlocks * 64, tail);
  }
}